// GCN_1365799600531
// MI455X (gfx1250) — compile-verified
//
#include <hip/hip_runtime.h>

typedef __attribute__((ext_vector_type(2))) float v2f;
typedef __attribute__((ext_vector_type(4))) float v4f;
typedef __attribute__((ext_vector_type(8))) float v8f;

#define IN_FT 128
#define OUT_FT 128
#define LDS_STRIDE 132   // 128 + 4 pad: B-frag lane reads hit distinct bank pairs

// Device-scope hardware f32 atomic add (no return -> STOREcnt, drained at endpgm).
// Inline asm guarantees global_atomic_add_f32 (no CAS-loop fallback), and
// scope:SCOPE_DEV makes the RMW coherent across all WGPs (default scope is WGP).
__device__ __forceinline__ void atomic_add_f32_dev(float* p, float v)
{
    asm volatile("global_atomic_add_f32 %0, %1, off scope:SCOPE_DEV"
                 :: "v"(p), "v"(v) : "memory");
}

// ---------------------------------------------------------------------------
// Kernel 1: fts[n,o] = sum_f seq[n,f] * W[o,f]  via V_WMMA_F32_16X16X4_F32
// Block = 256 threads = 8 waves; each wave does a 16-node tile x all 128 outs.
// ---------------------------------------------------------------------------
__global__ __launch_bounds__(256) void gcn_linear_wmma(
    const float* __restrict__ seq,   // [n_nodes, 128]
    const float* __restrict__ W,     // [128, 128] row-major [out][in]
    float* __restrict__ fts,         // [n_nodes, 128]
    int n_nodes)
{
    __shared__ float ldsW[OUT_FT * LDS_STRIDE];

    // Stage W into LDS (padded rows). 16384 floats / 256 threads = 16 float4 each.
    const v4f* W4 = (const v4f*)W;
    for (int i = threadIdx.x; i < (OUT_FT * IN_FT) / 4; i += 256) {
        int o = i >> 5;            // 32 float4 per row
        int f = (i & 31) << 2;
        *(v4f*)&ldsW[o * LDS_STRIDE + f] = W4[i];
    }
    __syncthreads();

    const int wave  = threadIdx.x >> 5;
    const int lane  = threadIdx.x & 31;
    const int m     = lane & 15;        // row-in-tile (A) / col-in-tile (B,C)
    const int khalf = lane >> 4;        // 0: K={0,1}, 1: K={2,3}
    const int koff  = khalf * 2;

    const int nodeBase = blockIdx.x * 128 + wave * 16;

    // A-fragment source row (clamped; clamped rows are never stored)
    int rowA = nodeBase + m;
    if (rowA >= n_nodes) rowA = n_nodes - 1;
    const float* aptr = seq + (size_t)rowA * IN_FT + koff;

    v8f acc[8] = {};   // 8 column tiles of 16 -> full 128 outputs

    #pragma unroll 4
    for (int kk = 0; kk < 32; ++kk) {          // K = 128 in steps of 4
        v2f a = *(const v2f*)(aptr + kk * 4);  // A[m, kk*4+koff .. +1]
        #pragma unroll
        for (int ot = 0; ot < 8; ++ot) {
            // B[k, n] = W[ot*16+n][k]; lane n holds K={koff,koff+1}
            v2f b = *(const v2f*)&ldsW[(ot * 16 + m) * LDS_STRIDE + kk * 4 + koff];
            acc[ot] = __builtin_amdgcn_wmma_f32_16x16x4_f32(
                false, a, false, b, (short)0, acc[ot], false, false);
        }
    }

    // C/D layout: lanes 0-15: VGPR v -> M=v; lanes 16-31: M=v+8; N=lane&15
    const int rowBase = nodeBase + 8 * khalf;
    #pragma unroll
    for (int ot = 0; ot < 8; ++ot) {
        #pragma unroll
        for (int v = 0; v < 8; ++v) {
            int row = rowBase + v;
            if (row < n_nodes)
                fts[(size_t)row * OUT_FT + ot * 16 + m] = acc[ot][v];
        }
    }
}

// ---------------------------------------------------------------------------
// Kernel 2: zero the accumulator (d_out) — 12.8M floats, float4 stores
// ---------------------------------------------------------------------------
__global__ void gcn_zero(float* __restrict__ out, int n4)
{
    int i = blockIdx.x * blockDim.x + threadIdx.x;
    if (i < n4) {
        v4f z = {};
        *(v4f*)(out + (size_t)i * 4) = z;
    }
}

// ---------------------------------------------------------------------------
// Kernel 3: scatter  out[dst] += val * fts[src]   (one wave per edge)
// 32 lanes x float4 = 128 features; fts row gathers hit the 192MB L2;
// 4 device-scope global_atomic_add_f32 per lane into the output row.
// ---------------------------------------------------------------------------
__global__ __launch_bounds__(256) void gcn_spmm_scatter(
    const float* __restrict__ fts,
    const float* __restrict__ edge_val,
    const int*  __restrict__ edge_src,
    const int*  __restrict__ edge_dst,
    float* __restrict__ out,
    int n_edges)
{
    int e = blockIdx.x * 8 + (threadIdx.x >> 5);
    if (e >= n_edges) return;
    int lane = threadIdx.x & 31;

    int   s = edge_src[e];
    int   d = edge_dst[e];
    float v = edge_val[e];

    v4f x = *(const v4f*)(fts + (size_t)s * OUT_FT + lane * 4);
    float* orow = out + (size_t)d * OUT_FT + lane * 4;

    atomic_add_f32_dev(orow + 0, x.x * v);
    atomic_add_f32_dev(orow + 1, x.y * v);
    atomic_add_f32_dev(orow + 2, x.z * v);
    atomic_add_f32_dev(orow + 3, x.w * v);
}

// ---------------------------------------------------------------------------
// Kernel 4: out = PReLU(out + bias)  (in place, one thread per element)
// ---------------------------------------------------------------------------
__global__ void gcn_bias_prelu(float* __restrict__ out,
                               const float* __restrict__ bias,
                               const float* __restrict__ prelu_a,
                               int n)
{
    int i = blockIdx.x * blockDim.x + threadIdx.x;
    if (i >= n) return;
    float x = out[i] + bias[i & (OUT_FT - 1)];
    float a = prelu_a[0];
    out[i] = (x >= 0.0f) ? x : a * x;
}

// ---------------------------------------------------------------------------
extern "C" void kernel_launch(void* const* d_in, const int* in_sizes, int n_in,
                              void* d_out, int out_size, void* d_ws, size_t ws_size,
                              hipStream_t stream)
{
    const float* seq      = (const float*)d_in[0];
    const float* W        = (const float*)d_in[1];
    const float* bias     = (const float*)d_in[2];
    const float* prelu_a  = (const float*)d_in[3];
    const float* edge_val = (const float*)d_in[4];
    const int*   edge_src = (const int*)  d_in[5];
    const int*   edge_dst = (const int*)  d_in[6];

    float* out = (float*)d_out;
    float* fts = (float*)d_ws;                 // 100000*128*4 = 51.2 MB scratch

    const int n_nodes = in_sizes[0] / IN_FT;   // 100000
    const int n_edges = in_sizes[4];           // 1600000
    const int n_elems = n_nodes * OUT_FT;

    gcn_linear_wmma<<<(n_nodes + 127) / 128, 256, 0, stream>>>(seq, W, fts, n_nodes);
    gcn_zero<<<((n_elems / 4) + 255) / 256, 256, 0, stream>>>(out, n_elems / 4);
    gcn_spmm_scatter<<<(n_edges + 7) / 8, 256, 0, stream>>>(
        fts, edge_val, edge_src, edge_dst, out, n_edges);
    gcn_bias_prelu<<<(n_elems + 255) / 256, 256, 0, stream>>>(out, bias, prelu_a, n_elems);
}